// TritonKeepMaskBuilder_80994493268396
// MI455X (gfx1250) — compile-verified
//
#include <hip/hip_runtime.h>
#include <hip/hip_bf16.h>

// KeepMaskBuilder for MI455X (gfx1250, wave32).
// One workgroup (1024 threads = 32 waves) per row of S=8192 tokens.
// HBM traffic is one read of gates+mask and one write of the mask: ~384MB
// total => ~16.5us at 23.3 TB/s. All sort/scan work happens in LDS.
//
// CDNA5-specific paths used:
//  - GLOBAL_LOAD_ASYNC_TO_LDS_B32 + s_wait_asynccnt for row staging
//  - V_WMMA_F32_16X16X4_F32 chains computing Y = L_tri * X (16-wide
//    inclusive segment scans on the matrix pipe, exact f32)

#define S_TOKENS   8192
#define BLOCK      1024
#define PER_THREAD (S_TOKENS / BLOCK) /* 8 */
#define NSEG       (S_TOKENS / 16)    /* 512 */
#define THRESH     0.9f

typedef __attribute__((ext_vector_type(2))) float v2f;
typedef __attribute__((ext_vector_type(8))) float v8f;

typedef __attribute__((address_space(1))) int g_i32;
typedef __attribute__((address_space(3))) int l_i32;

#if __has_builtin(__builtin_amdgcn_global_load_async_to_lds_b32)
#define HAVE_ASYNC_LDS 1
#else
#define HAVE_ASYNC_LDS 0
#endif

#if __has_builtin(__builtin_amdgcn_wmma_f32_16x16x4_f32)
#define HAVE_WMMA_F32X4 1
#else
#define HAVE_WMMA_F32X4 0
#endif

#if HAVE_ASYNC_LDS
__device__ __forceinline__ void wait_async_zero() {
#if __has_builtin(__builtin_amdgcn_s_wait_asynccnt)
  __builtin_amdgcn_s_wait_asynccnt(0);
#else
  asm volatile("s_wait_asynccnt 0" ::: "memory");
#endif
}
#endif

__global__ void __launch_bounds__(BLOCK)
keep_mask_kernel(const float* __restrict__ gates,
                 const int* __restrict__ amask,
                 float* __restrict__ out)
{
  __shared__ float sv[S_TOKENS];   // row buffer: gates -> shares -> sorted -> seg-scanned
  __shared__ float aux[NSEG];      // segment-sum scan
  __shared__ float wred[32];
  __shared__ int   ired[32];
  __shared__ float shTotal;
  __shared__ float shCut;
  __shared__ int   shK;

  const unsigned tid  = threadIdx.x;
  const unsigned lane = tid & 31u;
  const unsigned wave = tid >> 5;
  const unsigned row  = blockIdx.x;

  const float* grow = gates + (size_t)row * S_TOKENS;
  const int*   mrow = amask + (size_t)row * S_TOKENS;
  float*       orow = out   + (size_t)row * S_TOKENS;

  // ---------- Phase 0: stage gates row into LDS (async on CDNA5); mask -> regs
  int mreg[PER_THREAD];
#if HAVE_ASYNC_LDS
  #pragma unroll
  for (int t = 0; t < PER_THREAD; ++t) {
    unsigned p = tid + (unsigned)t * BLOCK;
    __builtin_amdgcn_global_load_async_to_lds_b32(
        (g_i32*)(grow + p),
        (l_i32*)(&sv[p]),
        0, 0);
  }
#else
  #pragma unroll
  for (int t = 0; t < PER_THREAD; ++t) {
    unsigned p = tid + (unsigned)t * BLOCK;
    sv[p] = grow[p];
  }
#endif
  #pragma unroll
  for (int t = 0; t < PER_THREAD; ++t)
    mreg[t] = mrow[tid + (unsigned)t * BLOCK];
#if HAVE_ASYNC_LDS
  wait_async_zero();
#endif
  __syncthreads();

  // ---------- Phase 1: gated values + row total (wave32 shuffle tree)
  float greg[PER_THREAD];
  float lsum = 0.0f;
  #pragma unroll
  for (int t = 0; t < PER_THREAD; ++t) {
    float g = sv[tid + (unsigned)t * BLOCK];
    g = mreg[t] ? g : 0.0f;
    greg[t] = g;
    lsum += g;
  }
  #pragma unroll
  for (int o = 16; o > 0; o >>= 1) lsum += __shfl_xor(lsum, o, 32);
  if (lane == 0) wred[wave] = lsum;
  __syncthreads();
  if (wave == 0) {
    float v = wred[lane];
    #pragma unroll
    for (int o = 16; o > 0; o >>= 1) v += __shfl_xor(v, o, 32);
    if (lane == 0) shTotal = v;
  }
  __syncthreads();
  const float inv = 1.0f / fmaxf(shTotal, 1e-12f);

  // ---------- Phase 2: shares into LDS (and keep in regs for final emit)
  float shreg[PER_THREAD];
  #pragma unroll
  for (int t = 0; t < PER_THREAD; ++t) {
    float s = greg[t] * inv;
    shreg[t] = s;
    sv[tid + (unsigned)t * BLOCK] = s;
  }

  // ---------- Phase 3: bitonic sort (descending) of 8192 shares in LDS
  for (unsigned k = 2; k <= S_TOKENS; k <<= 1) {
    for (unsigned j = k >> 1; j > 0; j >>= 1) {
      __syncthreads();
      #pragma unroll
      for (int t = 0; t < PER_THREAD; ++t) {
        unsigned i = tid + (unsigned)t * BLOCK;
        unsigned ixj = i ^ j;
        if (ixj > i) {
          float a = sv[i];
          float b = sv[ixj];
          bool up = ((i & k) == 0u);
          if ((a < b) == up) { sv[i] = b; sv[ixj] = a; }
        }
      }
    }
  }
  __syncthreads();

  // ---------- Phase 4a: 16-element inclusive segment scans via WMMA
  // Per wave: chunk of 256 sorted values = X (16x16, column c = segment c).
  // Y = L*X with L = 16x16 lower-triangular ones, decomposed into 4 chained
  // V_WMMA_F32_16X16X4_F32 ops: Y = sum_k L[:,4k:4k+4] * X[4k:4k+4,:].
  // Each lane stashes the pre-scan values it overwrites (for the cutoff).
  const unsigned hi   = lane >> 4;     // half-wave select
  const unsigned col  = lane & 15u;    // M for A-operand, N for B/D operands
  const unsigned base = wave * 256u;
  float stash[8];

#if HAVE_WMMA_F32X4
  v8f acc = {0.f, 0.f, 0.f, 0.f, 0.f, 0.f, 0.f, 0.f};
  #pragma unroll
  for (int kk = 0; kk < 4; ++kk) {
    v2f a, b;
    // A (16x4 f32): VGPR0 holds K = (hi?2:0), VGPR1 holds K+1; M = lane&15.
    const int c0 = 4 * kk + (hi ? 2 : 0);
    a.x = ((int)col >= c0    ) ? 1.0f : 0.0f;  // L[m][c0]
    a.y = ((int)col >= c0 + 1) ? 1.0f : 0.0f;  // L[m][c0+1]
    // B (4x16 f32): lanes 0-15 hold rows {0,1}, lanes 16-31 rows {2,3} of
    // the 4-row block (ISA B-operand convention: low lanes = low K half).
    const unsigned r0 = (unsigned)(4 * kk) + 2u * hi;
    b.x = sv[base + col * 16u + r0];
    b.y = sv[base + col * 16u + r0 + 1u];
    acc = __builtin_amdgcn_wmma_f32_16x16x4_f32(
        false, a, false, b, (short)0, acc, false, false);
  }
  #pragma unroll
  for (int v = 0; v < 8; ++v) {
    // D layout: lane l, VGPR v -> row M = v + 8*(l>=16), col N = l&15.
    unsigned p = base + col * 16u + (unsigned)v + 8u * hi;
    stash[v] = sv[p];
    sv[p] = acc[v];
  }
#else
  // Scalar fallback: both half-lanes of a column redundantly run the serial
  // scan; each writes (and stashes) only its own 8 rows.
  {
    float run = 0.0f;
    float res[8];
    #pragma unroll
    for (int r = 0; r < 16; ++r) {
      float x = sv[base + col * 16u + (unsigned)r];
      run += x;
      int rr = r - 8 * (int)hi;
      if (rr >= 0 && rr < 8) res[rr] = run;
    }
    #pragma unroll
    for (int v = 0; v < 8; ++v) {
      unsigned p = base + col * 16u + (unsigned)v + 8u * hi;
      stash[v] = sv[p];
      sv[p] = res[v];
    }
  }
#endif
  __syncthreads();

  // ---------- Phase 4b: Hillis-Steele scan of the 512 segment sums
  if (tid < NSEG) aux[tid] = sv[tid * 16u + 15u];
  __syncthreads();
  for (unsigned off = 1; off < NSEG; off <<= 1) {
    float cur = 0.0f, add = 0.0f;
    if (tid < NSEG) {
      cur = aux[tid];
      if (tid >= off) add = aux[tid - off];
    }
    __syncthreads();
    if (tid < NSEG) aux[tid] = cur + add;
    __syncthreads();
  }

  // ---------- Phase 5: K = |{p : cum[p] <= T}| (cum monotone), clamp >= 1
  int lcount = 0;
  #pragma unroll
  for (int t = 0; t < PER_THREAD; ++t) {
    unsigned p = tid + (unsigned)t * BLOCK;
    unsigned s = p >> 4;
    float cum = sv[p] + (s ? aux[s - 1u] : 0.0f);
    lcount += (cum <= THRESH) ? 1 : 0;
  }
  #pragma unroll
  for (int o = 16; o > 0; o >>= 1) lcount += __shfl_xor(lcount, o, 32);
  if (lane == 0) ired[wave] = lcount;
  __syncthreads();
  if (wave == 0) {
    int v = ired[lane];
    #pragma unroll
    for (int o = 16; o > 0; o >>= 1) v += __shfl_xor(v, o, 32);
    if (lane == 0) shK = (v < 1) ? 1 : v;
  }
  __syncthreads();

  // ---------- Phase 5b: broadcast cutoff = sorted[K-1] from its stash owner
  {
    const unsigned Km1 = (unsigned)(shK - 1);
    #pragma unroll
    for (int v = 0; v < 8; ++v) {
      unsigned p = base + col * 16u + (unsigned)v + 8u * hi;
      if (p == Km1) shCut = stash[v];
    }
  }
  __syncthreads();

  // ---------- Phase 6: emit keep mask in original token order from regs
  const float cut = shCut;
  #pragma unroll
  for (int t = 0; t < PER_THREAD; ++t) {
    unsigned p = tid + (unsigned)t * BLOCK;
    bool keep = (mreg[t] != 0) && (shreg[t] >= cut);
    orow[p] = keep ? 1.0f : 0.0f;
  }
}

extern "C" void kernel_launch(void* const* d_in, const int* in_sizes, int n_in,
                              void* d_out, int out_size, void* d_ws, size_t ws_size,
                              hipStream_t stream) {
  (void)n_in; (void)out_size; (void)d_ws; (void)ws_size;
  const float* gates = (const float*)d_in[0];
  const int*   amask = (const int*)d_in[1];
  float*       out   = (float*)d_out;
  const int rows = in_sizes[0] / S_TOKENS;  // 4096
  keep_mask_kernel<<<dim3(rows), dim3(BLOCK), 0, stream>>>(gates, amask, out);
}